// Linear_BC_69612829934079
// MI455X (gfx1250) — compile-verified
//
#include <hip/hip_runtime.h>

// out[b, f*64+n] = sum_{c,j} w[f,c,(n-j) & 63] * x[b, c*64+j]
// == GEMM D = X (256x4096) * W^T (4096x4096), W block-circulant.
// 64x64 circulant blocks generated on the fly from 64-float rows in LDS;
// GEMM runs on V_WMMA_F32_16X16X4_F32 (exact fp32, matches reference dtype).
//
// Block = 128 threads (4 wave32s) covering a 32(M) x 64(N == one 'f') tile.
// Each wave: one 16-wide N subtile, TWO 16-row M subtiles sharing the B
// fragment -> 2.0 LDS ops per WMMA instead of 3.0.
// Reduction: 64 chunks of 64 (each chunk == one 'c').

typedef __attribute__((ext_vector_type(2))) float v2f;
typedef __attribute__((ext_vector_type(4))) float v4f;
typedef __attribute__((ext_vector_type(8))) float v8f;

#define BM 32
#define XS_PITCH 68   // 64 + 4 pad: column reads hit 16 distinct banks (stride-4 mod 64)

__global__ __launch_bounds__(128)
void Linear_BC_circconv_wmma(const float* __restrict__ x,
                             const float* __restrict__ w,
                             float* __restrict__ out) {
  __shared__ float wsl[64 * 64];       // w[f][:][:] slice, 16 KB, loaded once
  __shared__ float xs[BM][XS_PITCH];   // staged x chunk (32 x 64, padded)

  const int tid   = threadIdx.x;       // 0..127
  const int lane  = tid & 31;
  const int wave  = tid >> 5;          // 0..3 == N subtile index

  const int f      = blockIdx.x;       // 0..63  -> output cols [f*64, f*64+64)
  const int mBlock = blockIdx.y * BM;  // 0..224 -> output rows

  // One-time load of the f-slice of weights (64x64 floats) into LDS.
  {
    const v4f* src = (const v4f*)(w + (size_t)f * 4096);
    v4f* dst = (v4f*)wsl;
#pragma unroll
    for (int i = 0; i < 8; ++i)
      dst[tid + i * 128] = src[tid + i * 128];
  }

  // WMMA f32 16x16x4 fragment coordinates (wave32):
  //   A: lane holds row m = lane%16; VGPR v holds k = (lane/16)*2 + v
  //   B: lane holds col n = lane%16; VGPR v holds k = (lane/16)*2 + v
  //   C/D: lane holds col n = lane%16; VGPR g holds row m = g + 8*(lane/16)
  const int mA    = lane & 15;
  const int kHalf = (lane >> 4) << 1;            // 0 or 2
  const int nL    = wave * 16 + (lane & 15);     // col within the 64-wide f block
  const int rowA0 = mA;                          // M subtile 0
  const int rowA1 = mA + 16;                     // M subtile 1

  v8f acc0 = {};
  v8f acc1 = {};

  for (int c = 0; c < 64; ++c) {
    __syncthreads();  // previous chunk's xs reads complete
    // Stage x[mBlock .. +31][c*64 .. +63]: 512 float4 loads across 128 threads.
#pragma unroll
    for (int i = 0; i < 4; ++i) {
      int q  = tid + i * 128;          // 0..511
      int m  = q >> 4;                 // 0..31
      int r4 = q & 15;                 // float4 slot within the 64-float row
      v4f v = *(const v4f*)(x + (size_t)(mBlock + m) * 4096 + c * 64 + r4 * 4);
      *(v4f*)&xs[m][r4 * 4] = v;
    }
    __syncthreads();

    const float* wrow = wsl + c * 64;  // circulant generator row for (f, c)
#pragma unroll
    for (int kk = 0; kk < 16; ++kk) {
      const int k0 = kk * 4 + kHalf;
      v2f a0 = *(const v2f*)&xs[rowA0][k0];        // ds_load_b64, 8B aligned
      v2f a1 = *(const v2f*)&xs[rowA1][k0];
      v2f b;                                       // shared by both M subtiles
      b.x = wrow[(nL - k0) & 63];                  // W^T[r][n] = w[f,c,(n-k) mod 64]
      b.y = wrow[(nL - k0 - 1) & 63];
      acc0 = __builtin_amdgcn_wmma_f32_16x16x4_f32(false, a0, false, b,
                                                   (short)0, acc0, false, false);
      acc1 = __builtin_amdgcn_wmma_f32_16x16x4_f32(false, a1, false, b,
                                                   (short)0, acc1, false, false);
    }
  }

  // Store D: coalesced 16-lane column runs.
  const int col     = f * 64 + nL;
  const int rowBase = mBlock + ((lane >> 4) << 3);
#pragma unroll
  for (int g = 0; g < 8; ++g) {
    out[(size_t)(rowBase + g) * 4096 + col]      = acc0[g];
    out[(size_t)(rowBase + 16 + g) * 4096 + col] = acc1[g];
  }
}

extern "C" void kernel_launch(void* const* d_in, const int* in_sizes, int n_in,
                              void* d_out, int out_size, void* d_ws, size_t ws_size,
                              hipStream_t stream) {
  (void)in_sizes; (void)n_in; (void)out_size; (void)d_ws; (void)ws_size;
  const float* x = (const float*)d_in[0];   // (256, 4096) fp32
  const float* w = (const float*)d_in[1];   // (64, 64, 64) fp32
  float* out = (float*)d_out;               // (256, 4096) fp32

  dim3 grid(64, 8);   // grid.x = f blocks (N), grid.y = batch blocks (M)
  Linear_BC_circconv_wmma<<<grid, 128, 0, stream>>>(x, w, out);
}